// ModelNew_50388556317008
// MI455X (gfx1250) — compile-verified
//
#include <hip/hip_runtime.h>
#include <hip/hip_bf16.h>

typedef __attribute__((ext_vector_type(2))) float v2f;
typedef __attribute__((ext_vector_type(8))) float v8f;

#define N_  16
#define CI_ 32
#define H_  64
#define W_  128
#define CO_ 64
#define OH_ 318
#define OW_ 638
#define PITCH 640  // LDS row pitch (floats)

// ---------------------------------------------------------------------------
// Kernel 1: zero-fill the dead output rows (oh % 5 in {0,2}); 128 per plane.
// grid = (planes=16*64, dead_row_idx=128), block = 256
// ---------------------------------------------------------------------------
__global__ __launch_bounds__(256) void convt_zero_dead(float* __restrict__ out) {
    const int plane = blockIdx.x;              // n*64 + co
    const int d     = blockIdx.y;              // 0..127
    const int oh    = 5 * (d >> 1) + 2 * (d & 1);
    float* row = out + ((size_t)plane * OH_ + oh) * OW_;
    #pragma unroll
    for (int ow = threadIdx.x; ow < OW_; ow += 256) row[ow] = 0.0f;
}

// ---------------------------------------------------------------------------
// Kernel 2: one block per (n, ih, kh, co-chunk) builds one full live output
// row slab (16 co x 638 ow) densely via WMMA f32 + LDS row assembly.
// grid = (16, 64, 3*4), block = 256 (8 waves; wave w owns iw in [16w,16w+16))
// ---------------------------------------------------------------------------
__global__ __launch_bounds__(256) void convt_rows(const float* __restrict__ x,
                                                  const float* __restrict__ w,
                                                  float* __restrict__ out) {
    const int n   = blockIdx.x;        // batch
    const int ih  = blockIdx.y;        // input row
    const int kz  = blockIdx.z;        // kh*4 + co_chunk
    const int kh  = kz >> 2;
    const int co4 = kz & 3;
    const int oh  = 5 * ih - 1 + 2 * kh;
    if (oh < 0 || oh >= OH_) return;   // boundary taps that fall off the image

    __shared__ float buf[16 * PITCH];  // 40 KB: 16 co rows, zero-padded gaps

    const int t = threadIdx.x;

    // zero LDS row buffer (b128)
    {
        float4* p = (float4*)buf;
        #pragma unroll
        for (int i = 0; i < 10; ++i)
            p[t + 256 * i] = make_float4(0.f, 0.f, 0.f, 0.f);
    }
    __syncthreads();

    const int wave = t >> 5;
    const int lane = t & 31;
    const int g    = lane >> 4;        // lane half (selects K pair / M+8)
    const int ln   = lane & 15;        // M (co) for A, N (iw) for B/D
    const int iw0  = wave * 16;
    const int co0  = co4 * 16;

    // ---- load B operand data: X[n, ci=0..31, ih, iw0+ln] into 16 regs ----
    // WMMA B 4x16 layout: reg v, lane -> B[K = 2*g + v, N = ln]
    float bx[16];
    const float* xb = x + (((size_t)n * CI_) * H_ + ih) * W_ + iw0 + ln;
    #pragma unroll
    for (int ks = 0; ks < 8; ++ks) {
        #pragma unroll
        for (int v = 0; v < 2; ++v) {
            const int ci = 4 * ks + 2 * g + v;
            bx[2 * ks + v] = xb[(size_t)ci * (H_ * W_)];
        }
    }

    #pragma unroll
    for (int kw = 0; kw < 3; ++kw) {
        v8f acc = {};
        #pragma unroll
        for (int ks = 0; ks < 8; ++ks) {
            // A 16x4 layout: reg v, lane -> A[M = ln, K = 2*g + v]
            // A[co', ci] = weight[ci, co0+co', kh, kw]   (layout (32,64,3,3))
            v2f a, b;
            #pragma unroll
            for (int v = 0; v < 2; ++v) {
                const int ci = 4 * ks + 2 * g + v;
                a[v] = w[((ci * CO_ + co0 + ln) * 3 + kh) * 3 + kw];
            }
            b[0] = bx[2 * ks];
            b[1] = bx[2 * ks + 1];
            // D = A x B + C  (16x16x4 fp32)
            acc = __builtin_amdgcn_wmma_f32_16x16x4_f32(
                false, a, false, b, (short)0, acc, false, false);
        }
        // D layout: reg r, lane -> D[M = r + 8*g, N = ln]
        // scatter into assembled row: ow = 5*iw + (2*kw - 1)
        const int ow = 5 * (iw0 + ln) + 2 * kw - 1;
        if (ow >= 0 && ow < OW_) {
            #pragma unroll
            for (int r = 0; r < 8; ++r)
                buf[(r + 8 * g) * PITCH + ow] = acc[r];
        }
    }
    __syncthreads();

    // ---- dense coalesced store of 16 complete output rows ----
    #pragma unroll
    for (int r = 0; r < 16; ++r) {
        float* orow = out + (((size_t)(n * CO_ + co0 + r) * OH_) + oh) * OW_;
        #pragma unroll
        for (int ow = t; ow < OW_; ow += 256)
            orow[ow] = buf[r * PITCH + ow];
    }
}

// ---------------------------------------------------------------------------
extern "C" void kernel_launch(void* const* d_in, const int* in_sizes, int n_in,
                              void* d_out, int out_size, void* d_ws, size_t ws_size,
                              hipStream_t stream) {
    const float* x  = (const float*)d_in[0];   // (16, 32, 64, 128)
    const float* wt = (const float*)d_in[1];   // (32, 64, 3, 3)
    float* out = (float*)d_out;                // (16, 64, 318, 638)

    // dead rows (oh%5 in {0,2}): 16*64 planes x 128 rows
    convt_zero_dead<<<dim3(N_ * CO_, 128, 1), 256, 0, stream>>>(out);
    // live rows: (n, ih, kh*4 + co_chunk)
    convt_rows<<<dim3(N_, H_, 3 * 4), 256, 0, stream>>>(x, wt, out);
}